// ComplexNN_88441966559494
// MI455X (gfx1250) — compile-verified
//
#include <hip/hip_runtime.h>
#include <cmath>

// ---------------------------------------------------------------------------
// MI455X (gfx1250) GEMM + bias + activation using V_WMMA_F32_16X16X4_F32
//
// 3-layer MLP, all fp32 (fp32 WMMA is RNE / denorm-preserving per CDNA5 ISA
// 7.12, so GEMM numerics match the fp32 reference):
//   h  = blend_act(x @ W1 + b1)        M=32768 N=1024 K=512
//   h2 = relu    (h @ W2 + b2)         M=32768 N=512  K=1024
//   y  =          h2 @ W3 + b3         M=32768 N=10   K=512
//
// Block: 256 threads (8 wave32), 128x128 output tile, waves in 4(M) x 2(N).
// Each wave owns a 32x64 sub-tile = 2x4 accumulators of 16x16 (64 acc VGPRs).
// Per k-step: 2 A frags + 4 B frags (6 ds_load_b64) feed 8 WMMAs
//   -> 192 B LDS traffic per WMMA (vs 384 B with a 16x32 wave tile),
//      keeping the 64-bank LDS (~256 B/clk) ahead of the matrix pipe.
// K staged through LDS in chunks of 32; A row-major, B transposed (NxK) so
// both fragment fetches are single ds_load_b64 of the {kk,kk+1}/{kk+2,kk+3}
// pairs required by the wave32 16x16x4 VGPR layout.
// ---------------------------------------------------------------------------

typedef float v2f __attribute__((ext_vector_type(2)));
typedef float v4f __attribute__((ext_vector_type(4)));
typedef float v8f __attribute__((ext_vector_type(8)));

#define TILE_M 128
#define TILE_N 128
#define KC     32
#define KPAD   (KC + 4)   // +4 dwords: 16B-aligned b128 stores, debanked rows

enum Act { ACT_NONE = 0, ACT_BLEND = 1, ACT_RELU = 2 };

__device__ __forceinline__ float blend_act_f(float x) {
    // x>0: 0.9*x + 0.1*tanh(x); else 0.5*tanh(x)
    float t = tanhf(x);
    return x > 0.0f ? fmaf(x, 0.9f, t * 0.1f) : t * 0.5f;
}

template <int ACT>
__global__ __launch_bounds__(256)
void gemm_bias_act(const float* __restrict__ A,   // M x K, row-major
                   const float* __restrict__ B,   // K x N, row-major
                   const float* __restrict__ bias,// N
                   float* __restrict__ C,         // M x N, row-major
                   int M, int N, int K)
{
    __shared__ float lA [TILE_M][KPAD];  // A tile, row-major  (128 x 36 dwords)
    __shared__ float lBt[TILE_N][KPAD];  // B tile, TRANSPOSED (n-major, k inner)

    const int tid  = threadIdx.x;
    const int lane = tid & 31;
    const int wave = tid >> 5;
    const int wm   = wave >> 1;        // 0..3 : which 32-row strip
    const int wn   = wave & 1;         // 0..1 : which 64-col strip
    const int half = lane >> 4;        // 0: lanes 0-15, 1: lanes 16-31
    const int l16  = lane & 15;

    const int m0 = blockIdx.y * TILE_M;
    const int n0 = blockIdx.x * TILE_N;
    const bool nFull = (n0 + TILE_N) <= N;   // uniform: fast vs guarded B path

    v8f acc[2][4];
    #pragma unroll
    for (int i = 0; i < 2; ++i)
        #pragma unroll
        for (int j = 0; j < 4; ++j)
            acc[i][j] = (v8f){0.f, 0.f, 0.f, 0.f, 0.f, 0.f, 0.f, 0.f};

    for (int k0 = 0; k0 < K; k0 += KC) {
        // ---- stage A tile: 128 rows x 32 floats = 1024 float4, 4 per thread
        #pragma unroll
        for (int i = 0; i < 4; ++i) {
            int slot = tid + i * 256;
            int r    = slot >> 3;          // 8 float4 per row
            int c4   = slot & 7;
            v4f v = *(const v4f*)(A + (size_t)(m0 + r) * K + k0 + c4 * 4);
            *(v4f*)(&lA[r][c4 * 4]) = v;   // stride 36 dwords -> 16B aligned
        }
        // ---- stage B tile transposed: 32 k-rows x 128 cols = 1024 float4
        #pragma unroll
        for (int i = 0; i < 4; ++i) {
            int slot = tid + i * 256;
            int k    = slot >> 5;          // 32 float4 per k-row
            int c4   = slot & 31;
            if (nFull) {
                v4f v = *(const v4f*)(B + (size_t)(k0 + k) * N + n0 + c4 * 4);
                lBt[c4 * 4 + 0][k] = v.x;
                lBt[c4 * 4 + 1][k] = v.y;
                lBt[c4 * 4 + 2][k] = v.z;
                lBt[c4 * 4 + 3][k] = v.w;
            } else {
                #pragma unroll
                for (int e = 0; e < 4; ++e) {
                    int n = n0 + c4 * 4 + e;
                    lBt[c4 * 4 + e][k] =
                        (n < N) ? B[(size_t)(k0 + k) * N + n] : 0.0f;
                }
            }
        }
        __syncthreads();

        // ---- 8 k-steps; each: 2 A frags + 4 B frags -> 8 WMMAs
        // A frag (16x4): lane -> row, k pair {kk+2*half, +1}
        // B frag (4x16): lane -> col, k pair {kk+2*half, +1}
        #pragma unroll
        for (int kk = 0; kk < KC; kk += 4) {
            const int ks = kk + 2 * half;
            v2f a[2], b[4];
            #pragma unroll
            for (int i = 0; i < 2; ++i)
                a[i] = *(const v2f*)(&lA[wm * 32 + i * 16 + l16][ks]);
            #pragma unroll
            for (int j = 0; j < 4; ++j)
                b[j] = *(const v2f*)(&lBt[wn * 64 + j * 16 + l16][ks]);
            #pragma unroll
            for (int i = 0; i < 2; ++i)
                #pragma unroll
                for (int j = 0; j < 4; ++j)
                    acc[i][j] = __builtin_amdgcn_wmma_f32_16x16x4_f32(
                        false, a[i], false, b[j], (short)0, acc[i][j],
                        false, false);
        }
        __syncthreads();
    }

    // ---- epilogue: C/D layout — VGPR r: row = base + r + 8*half, col = lane
    #pragma unroll
    for (int i = 0; i < 2; ++i) {
        const int row_base = m0 + wm * 32 + i * 16 + 8 * half;
        #pragma unroll
        for (int j = 0; j < 4; ++j) {
            const int col = n0 + wn * 64 + j * 16 + l16;
            if (col >= N) continue;
            const float bia = bias[col];
            #pragma unroll
            for (int r = 0; r < 8; ++r) {
                float v = acc[i][j][r] + bia;
                if constexpr (ACT == ACT_BLEND) v = blend_act_f(v);
                else if constexpr (ACT == ACT_RELU) v = fmaxf(v, 0.0f);
                C[(size_t)(row_base + r) * N + col] = v;
            }
        }
    }
}

extern "C" void kernel_launch(void* const* d_in, const int* in_sizes, int n_in,
                              void* d_out, int out_size, void* d_ws, size_t ws_size,
                              hipStream_t stream)
{
    (void)in_sizes; (void)n_in; (void)out_size; (void)ws_size;

    const float* x  = (const float*)d_in[0];   // 32768 x 512
    const float* W1 = (const float*)d_in[1];   // 512 x 1024
    const float* b1 = (const float*)d_in[2];   // 1024
    const float* W2 = (const float*)d_in[3];   // 1024 x 512
    const float* b2 = (const float*)d_in[4];   // 512
    const float* W3 = (const float*)d_in[5];   // 512 x 10
    const float* b3 = (const float*)d_in[6];   // 10
    float* out = (float*)d_out;                // 32768 x 10

    const int B = 32768, D = 512, H = 1024, H2 = 512, C = 10;

    float* h  = (float*)d_ws;                  // B*H  fp32 (128 MiB)
    float* h2 = h + (size_t)B * H;             // B*H2 fp32 ( 64 MiB)

    dim3 blk(256);
    // Same-stream launches serialize -> correct inter-layer ordering.
    gemm_bias_act<ACT_BLEND><<<dim3(H  / TILE_N,               B / TILE_M), blk, 0, stream>>>(
        x,  W1, b1, h,   B, H,  D);
    gemm_bias_act<ACT_RELU ><<<dim3(H2 / TILE_N,               B / TILE_M), blk, 0, stream>>>(
        h,  W2, b2, h2,  B, H2, H);
    gemm_bias_act<ACT_NONE ><<<dim3((C + TILE_N - 1) / TILE_N, B / TILE_M), blk, 0, stream>>>(
        h2, W3, b3, out, B, C,  H2);
}